// HeteroLinear_50508815401264
// MI455X (gfx1250) — compile-verified
//
#include <hip/hip_runtime.h>
#include <hip/hip_bf16.h>

typedef float v2f __attribute__((ext_vector_type(2)));
typedef float v8f __attribute__((ext_vector_type(8)));

#define NUM_TYPES 16
#define CH 64          // IN_CH == OUT_CH == 64
#define TILE_M 16
#define WAVES_PER_BLOCK 8

// ---------------- binning kernels ----------------

__global__ void zero_fill_kernel(int* counts, int* cursor, int* perm, int permLen) {
    int i = blockIdx.x * blockDim.x + threadIdx.x;
    if (blockIdx.x == 0 && threadIdx.x < NUM_TYPES) {
        counts[threadIdx.x] = 0;
        cursor[threadIdx.x] = 0;
    }
    for (int j = i; j < permLen; j += gridDim.x * blockDim.x)
        perm[j] = -1;
}

__global__ void hist_kernel(const int* __restrict__ type_vec, int n, int* counts) {
    __shared__ int lc[NUM_TYPES];
    if (threadIdx.x < NUM_TYPES) lc[threadIdx.x] = 0;
    __syncthreads();
    for (int i = blockIdx.x * blockDim.x + threadIdx.x; i < n; i += gridDim.x * blockDim.x)
        atomicAdd(&lc[type_vec[i] & (NUM_TYPES - 1)], 1);
    __syncthreads();
    if (threadIdx.x < NUM_TYPES) atomicAdd(&counts[threadIdx.x], lc[threadIdx.x]);
}

// exclusive prefix over 16 bucket counts, each bucket padded up to a multiple of 16
__global__ void prefix_kernel(const int* __restrict__ counts, int* offs) {
    if (threadIdx.x == 0) {
        int acc = 0;
        for (int t = 0; t < NUM_TYPES; ++t) {
            offs[t] = acc;
            acc += (counts[t] + (TILE_M - 1)) & ~(TILE_M - 1);
        }
        offs[NUM_TYPES] = acc;
    }
}

__global__ void scatter_kernel(const int* __restrict__ type_vec, int n,
                               const int* __restrict__ offs, int* cursor,
                               int* __restrict__ perm) {
    __shared__ int lc[NUM_TYPES];
    __shared__ int lbase[NUM_TYPES];
    if (threadIdx.x < NUM_TYPES) lc[threadIdx.x] = 0;
    __syncthreads();
    int i = blockIdx.x * blockDim.x + threadIdx.x;
    int t = 0, rank = 0;
    bool valid = (i < n);
    if (valid) {
        t = type_vec[i] & (NUM_TYPES - 1);
        rank = atomicAdd(&lc[t], 1);
    }
    __syncthreads();
    if (threadIdx.x < NUM_TYPES)
        lbase[threadIdx.x] = atomicAdd(&cursor[threadIdx.x], lc[threadIdx.x]);
    __syncthreads();
    if (valid)
        perm[offs[t] + lbase[t] + rank] = i;
}

// ---------------- WMMA GEMM kernel ----------------
// One wave32 per 16-row tile; all rows of a tile share one weight matrix.
// A: 16x64 gathered rows of x; B: W[t] 64x64; C: 16x64 (+bias), 4 N-tiles.
// V_WMMA_F32_16X16X4_F32, K-loop of 16 steps.

__global__ __launch_bounds__(WAVES_PER_BLOCK * 32)
void hetero_gemm_kernel(const float* __restrict__ x,
                        const float* __restrict__ w,
                        const float* __restrict__ bias,
                        const int* __restrict__ perm,
                        const int* __restrict__ offs,
                        float* __restrict__ out) {
    const int lane = threadIdx.x & 31;
    const int wave = threadIdx.x >> 5;
    const int tile = blockIdx.x * WAVES_PER_BLOCK + wave;
    const int rowBase = tile * TILE_M;

    const int totalPadded = offs[NUM_TYPES];
    if (rowBase >= totalPadded) return;

    // which type bucket does this tile belong to (buckets are 16-row aligned)
    int t = 0;
    #pragma unroll
    for (int i = 1; i < NUM_TYPES; ++i)
        if (rowBase >= offs[i]) t = i;

    const float* __restrict__ W = w + (size_t)t * CH * CH;
    const float* __restrict__ B = bias + (size_t)t * CH;

    const int h   = lane >> 4;   // lane half: selects K-pair (A) / K-pair (B)
    const int l15 = lane & 15;   // A: row M   ; B/C: column N

    // ---- gather A fragments: lane holds row m=l15, K pairs (4*kk + 2*h, +1)
    const int arow = perm[rowBase + l15];
    const float* __restrict__ xr = x + (size_t)(arow < 0 ? 0 : arow) * CH;
    const bool avalid = (arow >= 0);

    v2f afrag[16];
    #pragma unroll
    for (int kk = 0; kk < 16; ++kk) {
        if (avalid) {
            afrag[kk] = *(const v2f*)(xr + kk * 4 + 2 * h);
        } else {
            afrag[kk].x = 0.0f; afrag[kk].y = 0.0f;
        }
    }

    // ---- accumulators seeded with bias: every element a lane holds has N = nt*16 + l15
    v8f c[4];
    #pragma unroll
    for (int nt = 0; nt < 4; ++nt) {
        float bv = B[nt * 16 + l15];
        #pragma unroll
        for (int r = 0; r < 8; ++r) c[nt][r] = bv;
    }

    // ---- K loop: B fragment lane holds W[k0][n], W[k0+1][n], n = nt*16+l15
    #pragma unroll
    for (int nt = 0; nt < 4; ++nt) {
        #pragma unroll
        for (int kk = 0; kk < 16; ++kk) {
            const int k0 = kk * 4 + 2 * h;
            v2f b;
            b.x = W[(size_t)(k0 + 0) * CH + nt * 16 + l15];
            b.y = W[(size_t)(k0 + 1) * CH + nt * 16 + l15];
            c[nt] = __builtin_amdgcn_wmma_f32_16x16x4_f32(
                false, afrag[kk], false, b, (short)0, c[nt], false, false);
        }
    }

    // ---- scatter C: VGPR r, lane -> (M = r + 8*h, N = nt*16 + l15)
    #pragma unroll
    for (int r = 0; r < 8; ++r) {
        const int m = r + 8 * h;
        const int orow = perm[rowBase + m];
        if (orow >= 0) {
            float* op = out + (size_t)orow * CH + l15;
            op[0]  = c[0][r];
            op[16] = c[1][r];
            op[32] = c[2][r];
            op[48] = c[3][r];
        }
    }
}

// ---------------- launch ----------------

extern "C" void kernel_launch(void* const* d_in, const int* in_sizes, int n_in,
                              void* d_out, int out_size, void* d_ws, size_t ws_size,
                              hipStream_t stream) {
    const float* x        = (const float*)d_in[0];
    const float* weight   = (const float*)d_in[1];
    const float* bias     = (const float*)d_in[2];
    const int*   type_vec = (const int*)d_in[3];
    float*       out      = (float*)d_out;

    const int N = in_sizes[3];   // one type per row

    // workspace layout (ints): counts[16] | cursor[16] | offs[17] | pad | perm[N + 256]
    int* ws     = (int*)d_ws;
    int* counts = ws;
    int* cursor = ws + 16;
    int* offs   = ws + 32;
    int* perm   = ws + 64;
    const int permLen = N + NUM_TYPES * TILE_M;

    zero_fill_kernel<<<(permLen + 255) / 256, 256, 0, stream>>>(counts, cursor, perm, permLen);
    hist_kernel<<<1024, 256, 0, stream>>>(type_vec, N, counts);
    prefix_kernel<<<1, 32, 0, stream>>>(counts, offs);
    scatter_kernel<<<(N + 255) / 256, 256, 0, stream>>>(type_vec, N, offs, cursor, perm);

    const int maxTiles  = (N + TILE_M - 1) / TILE_M + NUM_TYPES;
    const int gemmBlocks = (maxTiles + WAVES_PER_BLOCK - 1) / WAVES_PER_BLOCK;
    hetero_gemm_kernel<<<gemmBlocks, WAVES_PER_BLOCK * 32, 0, stream>>>(
        x, weight, bias, perm, offs, out);
}